// LNKillingRelu_30760555774228
// MI455X (gfx1250) — compile-verified
//
#include <hip/hip_runtime.h>

typedef __attribute__((ext_vector_type(2))) float v2f;
typedef __attribute__((ext_vector_type(8))) float v8f;

#define FCH   64            // channels
#define LD    8             // LDIM
#define NVEC  4096          // N
#define NT    16            // n-tile per block
#define NCOL  (LD * NT)     // 128 GEMM columns (k,nn)
#define WSTR  160           // fpair row stride (floats); 160 % 64 == 32 -> conflict-free A b64 loads

// out[b,f,k,n] = (kf<0) ? x : x + kf*d,  d = W @ x (over channels),
// kf = x^T K_killing d over the LDIM axis (hardcoded sparse 8x8 form).
__global__ __launch_bounds__(256)
void lnkr_kernel(const float* __restrict__ x,
                 const float* __restrict__ W,
                 float* __restrict__ out)
{
    // W in f-pair-interleaved WMMA A-operand layout:
    //   wl[fpair*WSTR + g*2 + (f&1)] = W[g][f],  fpair = f>>1
    __shared__ __align__(16) float wl[(FCH / 2) * WSTR];   // 20 KB
    __shared__ __align__(16) float dl[FCH * NCOL];         // d[g][c], c = k*16+nn; 32 KB

    const int tid  = threadIdx.x;
    const int lane = tid & 31;
    const int hl   = lane >> 4;        // half-wave select (K split of WMMA operands)
    const int l16  = lane & 15;
    const int wave = tid >> 5;         // 8 waves

    const int b  = blockIdx.x >> 8;            // 16 batches
    const int n0 = (blockIdx.x & 255) * NT;    // 256 n-blocks

    const float* xb = x   + (size_t)b * FCH * LD * NVEC;
    float*       ob = out + (size_t)b * FCH * LD * NVEC;

    // ---- stage W into LDS in paired A-operand layout (coalesced global read) ----
#pragma unroll
    for (int t = 0; t < 16; ++t) {
        int idx = tid + t * 256;                    // idx = g*64 + f
        int g = idx >> 6, f = idx & 63;
        wl[(f >> 1) * WSTR + g * 2 + (f & 1)] = W[idx];
    }
    __syncthreads();

    // ---- phase 1: D = W * X  via v_wmma_f32_16x16x4_f32 ----
    {
        const int kq = wave;           // this wave's k value (column tile)
        v8f acc[4] = {};               // 4 M-tiles (g = m*16 .. m*16+15)
#pragma unroll
        for (int kk = 0; kk < 16; ++kk) {
            const int f0  = kk * 4 + hl * 2;
            const int fpA = kk * 2 + hl;
            // B tile (4 x 16): rows f0, f0+1 for this half-wave, col = n0+l16
            const float* px = xb + ((size_t)f0 * LD + kq) * NVEC + n0 + l16;
            v2f bm;
            bm.x = px[0];
            bm.y = px[(size_t)LD * NVEC];
#pragma unroll
            for (int m = 0; m < 4; ++m) {
                // A operand: single conflict-free ds_load_b64
                v2f am = *(const v2f*)&wl[fpA * WSTR + (m * 16 + l16) * 2];
                acc[m] = __builtin_amdgcn_wmma_f32_16x16x4_f32(
                    /*neg_a=*/false, am, /*neg_b=*/false, bm,
                    /*c_mod=*/(short)0, acc[m],
                    /*reuse_a=*/false, /*reuse_b=*/false);
            }
        }
        // spill D tiles to LDS: D row M = m*16 + hl*8 + r, col = kq*16 + l16
        const int c = kq * 16 + l16;
#pragma unroll
        for (int m = 0; m < 4; ++m) {
#pragma unroll
            for (int r = 0; r < 8; ++r) {
                dl[(m * 16 + hl * 8 + r) * NCOL + c] = acc[m][r];
            }
        }
    }
    __syncthreads();

    // ---- phase 2: Killing form + fused gated output ----
    // K = [[......1.],[.......1],[...1....],[..1.....],
    //      [....21..],[....12..],[1.......],[.1......]]
#pragma unroll
    for (int t = 0; t < 4; ++t) {
        const int p  = tid + t * 256;      // 1024 (f,nn) pairs
        const int f  = p >> 4;
        const int nn = p & 15;
        const float* xr = xb + (size_t)f * LD * NVEC + n0 + nn;
        float xv[LD], dv[LD];
#pragma unroll
        for (int k = 0; k < LD; ++k) {
            xv[k] = xr[(size_t)k * NVEC];          // cache-hot re-read of x
            dv[k] = dl[f * NCOL + k * 16 + nn];
        }
        float kf = xv[0] * dv[6] + xv[1] * dv[7]
                 + xv[2] * dv[3] + xv[3] * dv[2]
                 + xv[4] * (2.0f * dv[4] + dv[5])
                 + xv[5] * (dv[4] + 2.0f * dv[5])
                 + xv[6] * dv[0] + xv[7] * dv[1];
        // branchless gate: kf<0 -> x, else x + kf*d
        const float kfp = fmaxf(kf, 0.0f);
        float* orow = ob + (size_t)f * LD * NVEC + n0 + nn;
#pragma unroll
        for (int k = 0; k < LD; ++k) {
            orow[(size_t)k * NVEC] = fmaf(kfp, dv[k], xv[k]);
        }
    }
}

extern "C" void kernel_launch(void* const* d_in, const int* in_sizes, int n_in,
                              void* d_out, int out_size, void* d_ws, size_t ws_size,
                              hipStream_t stream) {
    (void)in_sizes; (void)n_in; (void)d_ws; (void)ws_size; (void)out_size;
    const float* x = (const float*)d_in[0];   // (16, 64, 8, 4096) fp32
    const float* W = (const float*)d_in[1];   // (64, 64) fp32
    float* out = (float*)d_out;               // (16, 64, 8, 4096) fp32

    // grid: 16 batches x 256 n-blocks of 16
    lnkr_kernel<<<dim3(16 * 256), dim3(256), 0, stream>>>(x, W, out);
}